// ModulatedConv2d_45217415692753
// MI455X (gfx1250) — compile-verified
//
#include <hip/hip_runtime.h>
#include <hip/hip_bf16.h>

typedef float v2f __attribute__((ext_vector_type(2)));
typedef float v8f __attribute__((ext_vector_type(8)));

__device__ __forceinline__ v8f wmma4(v2f a, v2f b, v8f c) {
    // D = A(16x4,f32) * B(4x16,f32) + C(16x16,f32)
    return __builtin_amdgcn_wmma_f32_16x16x4_f32(
        /*neg_a=*/false, a, /*neg_b=*/false, b,
        /*c_mod=*/(short)0, c, /*reuse_a=*/false, /*reuse_b=*/false);
}

#define CIN   256
#define COUT  256
#define HW    64
#define K9    2304   // CIN * 9
#define SDIM  512
#define NB    8

// ---------------------------------------------------------------------------
// Kernel 1: s[b,ic] = style[b] . mod_w[ic] + mod_b[ic], replicated 9x -> s9
// ---------------------------------------------------------------------------
__global__ __launch_bounds__(256) void style_s9_kernel(
    const float* __restrict__ style, const float* __restrict__ mod_w,
    const float* __restrict__ mod_b, float* __restrict__ s9)
{
    __shared__ float st[SDIM];
    const int b = blockIdx.x;
    const int tid = threadIdx.x;
    st[tid]       = style[b * SDIM + tid];
    st[tid + 256] = style[b * SDIM + tid + 256];
    __syncthreads();

    const float* wrow = mod_w + (size_t)tid * SDIM;
    float acc = mod_b[tid];
#pragma unroll 8
    for (int j = 0; j < SDIM; ++j) acc += st[j] * wrow[j];

#pragma unroll
    for (int t = 0; t < 9; ++t) s9[b * K9 + tid * 9 + t] = acc;
}

// ---------------------------------------------------------------------------
// Kernel 2: demod[b,oc] = rsqrt( sum_k (weight[oc,k]*s9[b,k])^2 + eps )
// ---------------------------------------------------------------------------
__global__ __launch_bounds__(256) void demod_kernel(
    const float* __restrict__ weight, const float* __restrict__ s9,
    float* __restrict__ demod)
{
    const int row = blockIdx.x;          // b*256 + oc
    const int b = row >> 8, oc = row & 255;
    const int tid = threadIdx.x;
    const float* wrow = weight + (size_t)oc * K9;
    const float* srow = s9 + (size_t)b * K9;

    float acc = 0.f;
    for (int k = tid; k < K9; k += 256) {
        float v = wrow[k] * srow[k];
        acc += v * v;
    }
    __shared__ float red[256];
    red[tid] = acc;
    __syncthreads();
#pragma unroll
    for (int s = 128; s > 0; s >>= 1) {
        if (tid < s) red[tid] += red[tid + s];
        __syncthreads();
    }
    if (tid == 0) demod[row] = rsqrtf(red[0] + 1e-8f);
}

// ---------------------------------------------------------------------------
// Kernel 3: implicit-GEMM conv with fp32 WMMA (16x16x4).
//   block tile: M=256 (all oc) x N=64 (one image row), K chunked by 16
//   wave tile : 64x32  (4x2 of 16x16 fragments, 8 acc chains)
//   A tile row-major [row][k] stride 20 ; B tile col-major [col][k] stride 20
// ---------------------------------------------------------------------------
#define LDA 20
#define LDBC 20
#define NCHUNK (K9 / 16)   // 144

__global__ __launch_bounds__(256) void conv_wmma_kernel(
    const float* __restrict__ x, const float* __restrict__ weight,
    const float* __restrict__ bias, const float* __restrict__ s9,
    const float* __restrict__ demod, float* __restrict__ out)
{
    __shared__ float lA[2][256 * LDA];   // 20480 B each
    __shared__ float lB[2][64 * LDBC];   //  5120 B each

    const int tid = threadIdx.x;
    const int h   = blockIdx.x;          // output image row (N tile = full row)
    const int b   = blockIdx.y;

    const int lane = tid & 31;
    const int wid  = tid >> 5;
    const int wm   = (wid & 3) * 64;     // wave M offset (4 x 64 = 256)
    const int wn   = (wid >> 2) * 32;    // wave N offset (2 x 32 = 64)
    const int half = lane >> 4;          // K-half select per WMMA layout
    const int lr   = lane & 15;

    // B staging geometry: 16(k) x 64(col) elements; thread -> (kcb0 + 4i, colB)
    const int colB = tid & 63;
    const int kcb0 = tid >> 6;

    const float* __restrict__ s9b = s9 + (size_t)b * K9;
    const float* __restrict__ xb  = x + (size_t)b * CIN * HW * HW;

    float4 aReg[4];
    float  bReg[4];

    auto loadChunk = [&](int t) {
        const int k0g = t * 16;
        // A: 256 rows x 16 k, float4 along k (coalesced, 16B aligned)
#pragma unroll
        for (int i = 0; i < 4; ++i) {
            int id4 = tid + 256 * i;
            int row = id4 >> 2;
            int kc4 = (id4 & 3) << 2;
            aReg[i] = *(const float4*)(weight + (size_t)row * K9 + k0g + kc4);
        }
        // B: im2col gather for one image row, scaled by s9
#pragma unroll
        for (int i = 0; i < 4; ++i) {
            int kc = kcb0 + 4 * i;
            int kk = k0g + kc;
            int ic = kk / 9;
            int r9 = kk - ic * 9;
            int kh = r9 / 3;
            int kw = r9 - kh * 3;
            int ih = h + kh - 1;
            int iw = colB + kw - 1;
            float v = 0.f;
            if ((unsigned)ih < (unsigned)HW && (unsigned)iw < (unsigned)HW)
                v = xb[((size_t)ic * HW + ih) * HW + iw];
            bReg[i] = v * s9b[kk];
        }
        // prefetch weight slice for chunk t+2 into L2/L1 (global_prefetch_b8)
        if (t + 2 < NCHUNK) {
            int id4 = tid;
            int row = id4 >> 2;
            int kc4 = (id4 & 3) << 2;
            __builtin_prefetch(weight + (size_t)row * K9 + k0g + 32 + kc4, 0, 0);
        }
    };
    auto storeChunk = [&](int buf) {
#pragma unroll
        for (int i = 0; i < 4; ++i) {
            int id4 = tid + 256 * i;
            int row = id4 >> 2;
            int kc4 = (id4 & 3) << 2;
            *(float4*)&lA[buf][row * LDA + kc4] = aReg[i];  // (row*20+kc4)%4==0
        }
#pragma unroll
        for (int i = 0; i < 4; ++i) {
            int kc = kcb0 + 4 * i;
            lB[buf][colB * LDBC + kc] = bReg[i];            // column-major
        }
    };

    v8f acc[4][2];
#pragma unroll
    for (int mt = 0; mt < 4; ++mt)
#pragma unroll
        for (int nt = 0; nt < 2; ++nt)
            acc[mt][nt] = (v8f){};

    loadChunk(0);
    storeChunk(0);
    __syncthreads();

    for (int t = 0; t < NCHUNK; ++t) {
        const int cur = t & 1;
        if (t + 1 < NCHUNK) loadChunk(t + 1);   // overlap global with compute

#pragma unroll
        for (int k0 = 0; k0 < 16; k0 += 4) {
            const int ak = k0 + half * 2;
            // A fragments (16x4): lane lr = M row, half selects K{0,1}/{2,3}
            v2f af[4];
#pragma unroll
            for (int mt = 0; mt < 4; ++mt)
                af[mt] = *(const v2f*)&lA[cur][(wm + mt * 16 + lr) * LDA + ak];
            // B fragments (4x16): lane lr = N col; contiguous K pair -> one b64
            v2f bf[2];
#pragma unroll
            for (int nt = 0; nt < 2; ++nt)
                bf[nt] = *(const v2f*)&lB[cur][(wn + nt * 16 + lr) * LDBC + ak];

#pragma unroll
            for (int mt = 0; mt < 4; ++mt)
#pragma unroll
                for (int nt = 0; nt < 2; ++nt)
                    acc[mt][nt] = wmma4(af[mt], bf[nt], acc[mt][nt]);
        }

        if (t + 1 < NCHUNK) storeChunk(1 ^ cur); // other buffer: safe post-compute
        __syncthreads();
    }

    // Epilogue: demodulate + bias. C/D layout: VGPR v -> M = v + half*8, N = lr.
    const float* __restrict__ demodb = demod + b * COUT;
    float* __restrict__ outb = out + ((size_t)b * COUT * HW + h) * HW;
#pragma unroll
    for (int mt = 0; mt < 4; ++mt) {
#pragma unroll
        for (int v = 0; v < 8; ++v) {
            int oc = wm + mt * 16 + half * 8 + v;
            float d  = demodb[oc];
            float bb = bias[oc];
#pragma unroll
            for (int nt = 0; nt < 2; ++nt)
                outb[(size_t)oc * (HW * HW) + wn + nt * 16 + lr] =
                    acc[mt][nt][v] * d + bb;
        }
    }
}

// ---------------------------------------------------------------------------
extern "C" void kernel_launch(void* const* d_in, const int* in_sizes, int n_in,
                              void* d_out, int out_size, void* d_ws, size_t ws_size,
                              hipStream_t stream) {
    const float* x      = (const float*)d_in[0];  // [8,256,64,64]
    const float* style  = (const float*)d_in[1];  // [8,512]
    const float* weight = (const float*)d_in[2];  // [256,256,3,3]
    const float* bias   = (const float*)d_in[3];  // [256]
    const float* mod_w  = (const float*)d_in[4];  // [256,512]
    const float* mod_b  = (const float*)d_in[5];  // [256]
    float* out = (float*)d_out;                   // [8,256,64,64]

    float* s9    = (float*)d_ws;                  // [8,2304]
    float* demod = s9 + NB * K9;                  // [8,256]

    style_s9_kernel<<<NB, 256, 0, stream>>>(style, mod_w, mod_b, s9);
    demod_kernel<<<NB * COUT, 256, 0, stream>>>(weight, s9, demod);

    dim3 grid(HW, NB);                            // (64 rows, 8 samples)
    conv_wmma_kernel<<<grid, 256, 0, stream>>>(x, weight, bias, s9, demod, out);
}